// WeightedPathSummation_10144712753166
// MI455X (gfx1250) — compile-verified
//
#include <hip/hip_runtime.h>

// WeightedPathSummation for MI455X (gfx1250, wave32).
//
// Stage A: E[a, l*256 + 2v+p] = sum_w y[a,w] * w_l[w,v,p]
//          via chained V_WMMA_F32_16X16X4_F32 (exact: y is one-hot f32).
// Stage B: out[a, obase_l + m*128 + v] = x0*e0 + x1*e1, streaming B128,
//          branchless per-l specialized loops, non-temporal x/out traffic.

typedef __attribute__((ext_vector_type(2))) float v2f;
typedef __attribute__((ext_vector_type(4))) float v4f;
typedef __attribute__((ext_vector_type(8))) float v8f;

#define NW   64       // species w
#define IN1  10240    // per-atom x row:  256 + 768 + 2304 + 6912
#define OUT1 5120     // per-atom out row: 128 + 384 + 1152 + 3456

// Stage-B worker for one l-block. All shape constants are compile-time:
//   CPA = float4 chunks per atom = 32*3^L, ITERS = 16*CPA/256,
//   XBASE/OBASE = flat offsets of this l-block.
// GUARD=false: block known fully in-range -> no per-iteration bounds check.
template <int L, int CPA, int ITERS, int XBASE, int OBASE, bool GUARD>
__device__ __forceinline__ void apply_block(const float* __restrict__ x,
                                            float* __restrict__ out,
                                            const float* E,
                                            int atomBase, int N, int tid)
{
    #pragma unroll 1
    for (int i = 0; i < ITERS; ++i) {
        const int chunk = i * 256 + tid;          // [0, 16*CPA)
        const int a     = chunk / CPA;            // local atom 0..15 (shift / magic mul)
        const int r     = chunk - a * CPA;        // float4 chunk within l-block
        const int ga    = atomBase + a;
        if (!GUARD || ga < N) {
            // x offset   = XBASE + r*8   (m*256 + v*2 collapses to 8r)
            // out offset = OBASE + r*4
            // E offset   = L*256 + (r mod 32)*8
            const v4f* xp = (const v4f*)(x + (size_t)ga * IN1 + XBASE + r * 8);
            const v4f  x0 = __builtin_nontemporal_load(xp);
            const v4f  x1 = __builtin_nontemporal_load(xp + 1);
            const v4f* ep = (const v4f*)(E + a * 1024 + L * 256 + ((r & 31) << 3));
            const v4f  e0 = ep[0];
            const v4f  e1 = ep[1];
            v4f r4;
            r4.x = x0.x * e0.x + x0.y * e0.y;
            r4.y = x0.z * e0.z + x0.w * e0.w;
            r4.z = x1.x * e1.x + x1.y * e1.y;
            r4.w = x1.z * e1.z + x1.w * e1.w;
            __builtin_nontemporal_store(r4, (v4f*)(out + (size_t)ga * OUT1 + OBASE + r * 4));
        }
    }
}

template <bool GUARD>
__device__ __forceinline__ void stage_b(const float* __restrict__ x,
                                        float* __restrict__ out,
                                        const float* E,
                                        int atomBase, int N, int tid)
{
    //            L  CPA  ITERS XBASE OBASE
    apply_block<0,  32,   2,    0,    0,    GUARD>(x, out, E, atomBase, N, tid);
    apply_block<1,  96,   6,    256,  128,  GUARD>(x, out, E, atomBase, N, tid);
    apply_block<2,  288,  18,   1024, 512,  GUARD>(x, out, E, atomBase, N, tid);
    apply_block<3,  864,  54,   3328, 1664, GUARD>(x, out, E, atomBase, N, tid);
}

__global__ __launch_bounds__(256) void wps_kernel(
    const float* __restrict__ x,  const float* __restrict__ y,
    const float* __restrict__ w0, const float* __restrict__ w1,
    const float* __restrict__ w2, const float* __restrict__ w3,
    float* __restrict__ out, int N)
{
    // E tile for 16 atoms: 16 rows x 1024 effective-weight columns = 64 KB LDS.
    __shared__ float E[16 * 1024];

    const int tid      = threadIdx.x;
    const int wave     = tid >> 5;       // 0..7
    const int lane     = tid & 31;
    const int lane16   = lane & 15;
    const int laneHi   = lane >> 4;      // 0 or 1
    const int atomBase = blockIdx.x << 4;

    // ---------------- Stage A: E = Y(16x64) @ Wcat(64x1024) via WMMA ----------------
    // A fragment (16x4 f32, M x K): lane m = lane16;
    //   vgpr0 -> K = 4j + 2*laneHi, vgpr1 -> K = 4j + 2*laneHi + 1   (contiguous pair)
    int aIdx = atomBase + lane16;
    if (aIdx >= N) aIdx = N - 1;                     // clamp (N is a multiple of 16 anyway)
    const float* yrow = y + (size_t)aIdx * NW + 2 * laneHi;

    v2f afrag[16];
    #pragma unroll
    for (int j = 0; j < 16; ++j) {
        afrag[j] = *(const v2f*)(yrow + 4 * j);      // 8B-aligned float2
    }

    // Each wave owns 128 columns of E (8 tiles of 16). A wave's columns live
    // entirely in one l-block: l = wave >> 1 (t-invariant -> hoist the select).
    const int l = wave >> 1;
    const float* wl = (l == 0) ? w0 : (l == 1) ? w1 : (l == 2) ? w2 : w3;
    const int colInL0 = (wave & 1) * 128 + lane16;   // column within w_l, before +t*16
    // B fragment (4x16 f32, K x N): row K = 4j + vgpr + 2*laneHi, col = lane16.
    const float* wpBase = wl + colInL0 + (size_t)(2 * laneHi) * 256;
    const int colE0 = l * 256 + colInL0;             // E column, before +t*16

    #pragma unroll 1
    for (int t = 0; t < 8; ++t) {
        const float* wp = wpBase + t * 16;
        v8f acc = {};
        #pragma unroll
        for (int j = 0; j < 16; ++j) {               // K = 64 in steps of 4
            v2f b;
            b.x = wp[(size_t)(4 * j    ) * 256];
            b.y = wp[(size_t)(4 * j + 1) * 256];
            acc = __builtin_amdgcn_wmma_f32_16x16x4_f32(
                      false, afrag[j], false, b, (short)0, acc, false, false);
        }
        // C/D layout: row M = g + 8*laneHi, col = lane16 within tile.
        const int col = colE0 + t * 16;
        #pragma unroll
        for (int g = 0; g < 8; ++g) {
            E[(g + 8 * laneHi) * 1024 + col] = acc[g];
        }
    }
    __syncthreads();

    // ---------------- Stage B: stream x, apply E (per-l specialized) ----------------
    if (atomBase + 16 <= N) {
        stage_b<false>(x, out, E, atomBase, N, tid);  // fast path: no bounds checks
    } else {
        stage_b<true >(x, out, E, atomBase, N, tid);  // tail block (unused when N%16==0)
    }
}

extern "C" void kernel_launch(void* const* d_in, const int* in_sizes, int n_in,
                              void* d_out, int out_size, void* d_ws, size_t ws_size,
                              hipStream_t stream) {
    (void)n_in; (void)out_size; (void)d_ws; (void)ws_size;
    const float* x  = (const float*)d_in[0];
    const float* y  = (const float*)d_in[1];
    const float* w0 = (const float*)d_in[2];
    const float* w1 = (const float*)d_in[3];
    const float* w2 = (const float*)d_in[4];
    const float* w3 = (const float*)d_in[5];
    float* out = (float*)d_out;

    const int N = in_sizes[0] / IN1;                 // 20000
    const int blocks = (N + 15) / 16;                // 1250
    hipLaunchKernelGGL(wps_kernel, dim3(blocks), dim3(256), 0, stream,
                       x, y, w0, w1, w2, w3, out, N);
}